// GCN_67740224193041
// MI455X (gfx1250) — compile-verified
//
#include <hip/hip_runtime.h>

// ---------------------------------------------------------------------------
// 3-layer GCN on MI455X (gfx1250, wave32).
//   GEMMs:   V_WMMA_F32_16X16X4_F32 (fp32 matrix pipe, no precision loss),
//            A-slab staged to LDS via GLOBAL_LOAD_ASYNC_TO_LDS_B128
//            (ASYNCcnt + workgroup barrier), bank-conflict-free padded rows.
//   Self-loop term fused into GEMM epilogue (saves a full N*H pass/layer).
//   Scatter: one wave per edge, float4 per lane, global_atomic_add_f32
//            (51 MB destination buffer is L2-resident: 192 MB L2).
// ---------------------------------------------------------------------------

typedef __attribute__((ext_vector_type(2))) float v2f;
typedef __attribute__((ext_vector_type(8))) float v8f;

#define HID 128
#define LDSW 132  // padded LDS row stride (floats): bank = 4*row mod 64, conflict-free
#define DIV_UP(a, b) (((a) + (b) - 1) / (b))

// ---- degree / normalization -----------------------------------------------
__global__ void k_deg_init(float* __restrict__ deg, int n) {
  int i = blockIdx.x * blockDim.x + threadIdx.x;
  if (i < n) deg[i] = 1.0f;  // self-loop contribution
}

__global__ void k_deg_accum(const int* __restrict__ dst, float* __restrict__ deg, int e) {
  int i = blockIdx.x * blockDim.x + threadIdx.x;
  if (i < e) atomicAdd(&deg[dst[i]], 1.0f);
}

__global__ void k_dinv(float* __restrict__ deg, int n) {
  int i = blockIdx.x * blockDim.x + threadIdx.x;
  if (i < n) deg[i] = rsqrtf(deg[i]);  // deg >= 1 always (self loop)
}

// ---- layer-1 GEMM: [N,9] @ [9,128], self-loop term fused ------------------
__global__ __launch_bounds__(256) void k_gemm_in9(const float* __restrict__ x,
                                                  const float* __restrict__ W,
                                                  const float* __restrict__ dinv,
                                                  float* __restrict__ hw,
                                                  float* __restrict__ selfo, int n) {
  int t = blockIdx.x * blockDim.x + threadIdx.x;  // n*32 threads, 4 feats each
  if (t >= n * 32) return;
  int node = t >> 5;
  int f0 = (t & 31) * 4;
  const float* xr = x + node * 9;
  float a0 = 0.f, a1 = 0.f, a2 = 0.f, a3 = 0.f;
#pragma unroll
  for (int k = 0; k < 9; ++k) {
    float xv = xr[k];
    const float* wr = W + k * HID + f0;
    a0 += xv * wr[0]; a1 += xv * wr[1]; a2 += xv * wr[2]; a3 += xv * wr[3];
  }
  float4 o = {a0, a1, a2, a3};
  *(float4*)(hw + (size_t)node * HID + f0) = o;
  float dv = dinv[node];
  float c = dv * dv;
  float4 s = {a0 * c, a1 * c, a2 * c, a3 * c};
  *(float4*)(selfo + (size_t)node * HID + f0) = s;
}

// ---- hidden GEMM: [N,128] @ [128,128] via V_WMMA_F32_16X16X4_F32 ----------
// Block = 256 threads = 8 waves; one 16-row slab per block, one 16x16 output
// tile per wave. A slab (16x128) is copied to LDS with async-to-LDS b128
// (one row per wave-instruction), then fed to WMMA from LDS. W (L2-resident)
// is read directly. Self-loop term (dinv^2 * hw) fused into the epilogue.
// EXEC is all-ones throughout, as WMMA requires. N % 16 == 0.
__global__ __launch_bounds__(256) void k_gemm128_wmma(const float* __restrict__ A,
                                                      const float* __restrict__ W,
                                                      const float* __restrict__ dinv,
                                                      float* __restrict__ hw,
                                                      float* __restrict__ selfo) {
  __shared__ float sA[16 * LDSW];  // 16 rows, padded stride

  const int mtile = blockIdx.x;
  const int wv    = threadIdx.x >> 5;   // wave id 0..7 (also column tile)
  const int lane  = threadIdx.x & 31;
  const int r16   = lane & 15;
  const int khalf = lane >> 4;          // 0: K {k0,k0+1}, 1: K {k0+2,k0+3}

  // --- async copy A slab to LDS: wave w moves rows w and w+8 ---------------
  {
    unsigned l0 = (unsigned)(uintptr_t)&sA[wv * LDSW + lane * 4];
    const float* g0 = A + (size_t)(mtile * 16 + wv) * HID + lane * 4;
    unsigned l1 = (unsigned)(uintptr_t)&sA[(wv + 8) * LDSW + lane * 4];
    const float* g1 = g0 + 8 * HID;
    asm volatile("global_load_async_to_lds_b128 %0, %1, off\n\t"
                 "global_load_async_to_lds_b128 %2, %3, off"
                 :: "v"(l0), "v"(g0), "v"(l1), "v"(g1)
                 : "memory");
    asm volatile("s_wait_asynccnt 0" ::: "memory");
  }
  __syncthreads();

  const float* arow = &sA[r16 * LDSW];  // A row (M = r16) in LDS
  const int bcol = wv * 16 + r16;       // B col (N = r16)

  v8f acc = {};
#pragma unroll 8
  for (int k0 = 0; k0 < HID; k0 += 4) {
    const int k = k0 + 2 * khalf;
    v2f a = *(const v2f*)(arow + k);      // ds_load_b64: A[M][k], A[M][k+1]
    v2f b;
    b.x = W[(k + 0) * HID + bcol];        // B[k][N]
    b.y = W[(k + 1) * HID + bcol];        // B[k+1][N]
    // 8 args: (neg_a, A, neg_b, B, c_mod, C, reuse_a, reuse_b)
    acc = __builtin_amdgcn_wmma_f32_16x16x4_f32(false, a, false, b,
                                                (short)0, acc, false, false);
  }
  // D layout: lane group khalf holds rows 8*khalf + r (r=0..7), col = r16.
  const int row0 = mtile * 16 + 8 * khalf;
  float* o  = hw    + (size_t)row0 * HID + wv * 16 + r16;
  float* so = selfo + (size_t)row0 * HID + wv * 16 + r16;
#pragma unroll
  for (int r = 0; r < 8; ++r) {
    float v = acc[r];
    o[r * HID] = v;
    float dv = dinv[row0 + r];            // broadcast across half-wave
    so[r * HID] = v * dv * dv;
  }
}

// ---- aggregation: scatter-add over edges (one wave / edge, 4 floats / lane)
__global__ __launch_bounds__(256) void k_agg_edges(const int* __restrict__ src,
                                                   const int* __restrict__ dst,
                                                   const float* __restrict__ dinv,
                                                   const float* __restrict__ hw,
                                                   float* __restrict__ out, int e) {
  long long t = (long long)blockIdx.x * blockDim.x + threadIdx.x;
  long long edge = t >> 5;
  if (edge >= e) return;
  int lane = (int)(t & 31);
  int s = src[edge], d = dst[edge];
  float c = dinv[s] * dinv[d];
  const float* hs = hw + (size_t)s * HID + lane * 4;
  float* od = out + (size_t)d * HID + lane * 4;
  float4 v = *(const float4*)hs;
  atomicAdd(&od[0], v.x * c);
  atomicAdd(&od[1], v.y * c);
  atomicAdd(&od[2], v.z * c);
  atomicAdd(&od[3], v.w * c);
}

// ---- bias + optional ReLU --------------------------------------------------
__global__ __launch_bounds__(256) void k_bias_act(float* __restrict__ h,
                                                  const float* __restrict__ b,
                                                  int n, int do_relu) {
  int t = blockIdx.x * blockDim.x + threadIdx.x;
  if (t >= n * 32) return;
  int node = t >> 5;
  int f0 = (t & 31) * 4;
  float* p = h + (size_t)node * HID + f0;
  float4 v = *(const float4*)p;
  v.x += b[f0 + 0]; v.y += b[f0 + 1]; v.z += b[f0 + 2]; v.w += b[f0 + 3];
  if (do_relu) {
    v.x = fmaxf(v.x, 0.f); v.y = fmaxf(v.y, 0.f);
    v.z = fmaxf(v.z, 0.f); v.w = fmaxf(v.w, 0.f);
  }
  *(float4*)p = v;
}

// ---- pooling ---------------------------------------------------------------
__global__ void k_pool_init(float* __restrict__ pooled, float* __restrict__ counts,
                            int g) {
  int i = blockIdx.x * blockDim.x + threadIdx.x;
  if (i < g * HID) pooled[i] = 0.0f;
  if (i < g) counts[i] = 0.0f;
}

__global__ __launch_bounds__(256) void k_pool_accum(const float* __restrict__ h,
                                                    const int* __restrict__ batch,
                                                    float* __restrict__ pooled,
                                                    float* __restrict__ counts, int n) {
  int t = blockIdx.x * blockDim.x + threadIdx.x;
  if (t >= n * 32) return;
  int node = t >> 5;
  int lane = t & 31;
  int g = batch[node];
  float4 v = *(const float4*)(h + (size_t)node * HID + lane * 4);
  float* pg = pooled + g * HID + lane * 4;
  atomicAdd(&pg[0], v.x);
  atomicAdd(&pg[1], v.y);
  atomicAdd(&pg[2], v.z);
  atomicAdd(&pg[3], v.w);
  if (lane == 0) atomicAdd(&counts[g], 1.0f);
}

// ---- final linear: out[g] = (pooled[g] / count[g]) . lin_w + lin_b ---------
__global__ void k_final(const float* __restrict__ pooled,
                        const float* __restrict__ counts,
                        const float* __restrict__ lw, const float* __restrict__ lb,
                        float* __restrict__ out) {
  int g = blockIdx.x;
  int lane = threadIdx.x;  // 32 threads (one wave)
  float acc = 0.f;
#pragma unroll
  for (int f = lane; f < HID; f += 32) acc += pooled[g * HID + f] * lw[f];
#pragma unroll
  for (int off = 16; off > 0; off >>= 1) acc += __shfl_xor(acc, off, 32);
  if (lane == 0) out[g] = acc / fmaxf(counts[g], 1.0f) + lb[0];
}

// ---------------------------------------------------------------------------
extern "C" void kernel_launch(void* const* d_in, const int* in_sizes, int n_in,
                              void* d_out, int out_size, void* d_ws, size_t ws_size,
                              hipStream_t stream) {
  const float* x   = (const float*)d_in[0];
  const int*   ei  = (const int*)d_in[1];  // edge_index [2,E]
  const int*   bat = (const int*)d_in[2];  // batch [N]
  const float* W1  = (const float*)d_in[3];
  const float* b1  = (const float*)d_in[4];
  const float* W2  = (const float*)d_in[5];
  const float* b2  = (const float*)d_in[6];
  const float* W3  = (const float*)d_in[7];
  const float* b3  = (const float*)d_in[8];
  const float* lw  = (const float*)d_in[9];
  const float* lb  = (const float*)d_in[10];
  float* out = (float*)d_out;

  const int N = in_sizes[2];
  const int E = in_sizes[1] / 2;
  const int G = out_size;

  const int* src = ei;
  const int* dst = ei + E;

  // Workspace layout (~103 MB): dinv | bufA (hw) | bufB (h) | pooled | counts
  char* ws = (char*)d_ws;
  size_t off = 0;
  float* dinv = (float*)(ws + off);
  off += ((size_t)N * 4 + 255) & ~(size_t)255;
  float* bufA = (float*)(ws + off);
  off += (size_t)N * HID * 4;
  float* bufB = (float*)(ws + off);
  off += (size_t)N * HID * 4;
  float* pooled = (float*)(ws + off);
  off += (size_t)G * HID * 4;
  float* counts = (float*)(ws + off);

  const int B256 = 256;
  const int gN  = DIV_UP(N, B256);
  const int gE  = DIV_UP(E, B256);
  const int gNF = DIV_UP(N * 32, B256);                 // N*128 feats, 4/thread
  const int gEF = (int)DIV_UP((long long)E * 32, B256);
  const int gMT = N / 16;                               // WMMA row-slab tiles

  // normalization: deg -> dinv (in place)
  k_deg_init<<<gN, B256, 0, stream>>>(dinv, N);
  k_deg_accum<<<gE, B256, 0, stream>>>(dst, dinv, E);
  k_dinv<<<gN, B256, 0, stream>>>(dinv, N);

  // ---- layer 1: relu(agg(x @ W1) + b1) ----
  k_gemm_in9<<<gNF, B256, 0, stream>>>(x, W1, dinv, bufA, bufB, N);
  k_agg_edges<<<gEF, B256, 0, stream>>>(src, dst, dinv, bufA, bufB, E);
  k_bias_act<<<gNF, B256, 0, stream>>>(bufB, b1, N, 1);

  // ---- layer 2: relu(agg(h @ W2) + b2) ----
  k_gemm128_wmma<<<gMT, B256, 0, stream>>>(bufB, W2, dinv, bufA, bufB);
  k_agg_edges<<<gEF, B256, 0, stream>>>(src, dst, dinv, bufA, bufB, E);
  k_bias_act<<<gNF, B256, 0, stream>>>(bufB, b2, N, 1);

  // ---- layer 3: agg(h @ W3) + b3 (no relu) ----
  k_gemm128_wmma<<<gMT, B256, 0, stream>>>(bufB, W3, dinv, bufA, bufB);
  k_agg_edges<<<gEF, B256, 0, stream>>>(src, dst, dinv, bufA, bufB, E);
  k_bias_act<<<gNF, B256, 0, stream>>>(bufB, b3, N, 0);

  // ---- mean pool + final linear ----
  k_pool_init<<<DIV_UP(G * HID, B256), B256, 0, stream>>>(pooled, counts, G);
  k_pool_accum<<<gNF, B256, 0, stream>>>(bufB, bat, pooled, counts, N);
  k_final<<<G, 32, 0, stream>>>(pooled, counts, lw, lb, out);

  (void)n_in; (void)ws_size;
}